// Entmax_33234456936884
// MI455X (gfx1250) — compile-verified
//
#include <hip/hip_runtime.h>
#include <math.h>

// entmax-1.5 bisection, one wave32 per row of 2048 f32.
// alpha = 1.5  =>  am1 = 0.5, 1/am1 = 2 (pow == square)
// hi = max(am1z) == 0 exactly (row max subtracted), lo = -1.
// Candidates: am1z > -1 (elements <= -1 contribute 0 to every bisection sum
// and to the final p, since tau in (-1, 0)). For N(0,1) rows ~55/2048 survive,
// cutting the 50-iteration loop by ~30x vs scanning the full row.
//
// WMMA deliberately not used: the inner op is a thresholded sum-of-squares
// (sub+clamp on VALU per element is unavoidable); a diff*diff^T WMMA would
// need dual-layout marshalling per iteration and uses only the diagonal of
// the 16x16 output -- strictly worse than v_fmac.

#define ROWLEN          2048
#define Q_PER_LANE      16                // 16 x float4 = 64 floats per lane
#define WAVES_PER_BLOCK 4
#define NITER           50
#define PADVAL          (-4.0f)           // max(PADVAL - mid, 0) == 0 for mid in [-1,0]

__device__ __forceinline__ float wave_sum(float v) {
    v += __shfl_xor(v, 1, 32);
    v += __shfl_xor(v, 2, 32);
    v += __shfl_xor(v, 4, 32);
    v += __shfl_xor(v, 8, 32);
    v += __shfl_xor(v, 16, 32);
    return v;
}

__device__ __forceinline__ float wave_max(float v) {
    v = fmaxf(v, __shfl_xor(v, 1, 32));
    v = fmaxf(v, __shfl_xor(v, 2, 32));
    v = fmaxf(v, __shfl_xor(v, 4, 32));
    v = fmaxf(v, __shfl_xor(v, 8, 32));
    v = fmaxf(v, __shfl_xor(v, 16, 32));
    return v;
}

// 50 bisection iterations + normalization sum, specialized on the number of
// register-resident candidate chunks (padded/32). Returns {tau, 1/max(s,1e-12)}.
template <int NC, bool TAIL>
__device__ __forceinline__ float2 bisect_row(const float* __restrict__ my,
                                             float r0, float r1, float r2, float r3,
                                             int padded, int lane) {
    float lo = -1.0f, hi = 0.0f;
#pragma unroll 1
    for (int it = 0; it < NITER; ++it) {
        const float mid = 0.5f * (lo + hi);
        float d0 = fmaxf(r0 - mid, 0.0f);
        float p  = d0 * d0;
        if (NC > 1) { float d = fmaxf(r1 - mid, 0.0f); p = fmaf(d, d, p); }
        if (NC > 2) { float d = fmaxf(r2 - mid, 0.0f); p = fmaf(d, d, p); }
        if (NC > 3) { float d = fmaxf(r3 - mid, 0.0f); p = fmaf(d, d, p); }
        if (TAIL) {
            for (int idx = 128 + lane; idx < padded; idx += 32) {   // rare overflow path
                float c = my[idx];
                float d = fmaxf(c - mid, 0.0f);
                p = fmaf(d, d, p);
            }
        }
        const float s = wave_sum(p);
        if (s > 1.0f) lo = mid; else hi = mid;
    }
    const float tau = 0.5f * (lo + hi);

    // normalization sum over candidates (non-candidates contribute exactly 0)
    float d0 = fmaxf(r0 - tau, 0.0f);
    float p  = d0 * d0;
    if (NC > 1) { float d = fmaxf(r1 - tau, 0.0f); p = fmaf(d, d, p); }
    if (NC > 2) { float d = fmaxf(r2 - tau, 0.0f); p = fmaf(d, d, p); }
    if (NC > 3) { float d = fmaxf(r3 - tau, 0.0f); p = fmaf(d, d, p); }
    if (TAIL) {
        for (int idx = 128 + lane; idx < padded; idx += 32) {
            float c = my[idx];
            float d = fmaxf(c - tau, 0.0f);
            p = fmaf(d, d, p);
        }
    }
    const float s = wave_sum(p);
    return make_float2(tau, 1.0f / fmaxf(s, 1e-12f));
}

__global__ __launch_bounds__(32 * WAVES_PER_BLOCK)
void entmax15_rows_kernel(const float* __restrict__ x,
                          float* __restrict__ out,
                          int nrows) {
    __shared__ float cand_all[WAVES_PER_BLOCK][ROWLEN + 32];

    const int lane = threadIdx.x & 31;
    const int wave = threadIdx.x >> 5;
    int row = blockIdx.x * WAVES_PER_BLOCK + wave;
    if (row >= nrows) row = nrows - 1;   // duplicate last row; deterministic, keeps barriers uniform

    float* __restrict__ my = cand_all[wave];
    const int trash = ROWLEN + lane;     // per-lane scratch slot, never read

    const float4* __restrict__ xr = (const float4*)(x + (size_t)row * ROWLEN);
    float4* __restrict__ orow     = (float4*)(out + (size_t)row * ROWLEN);

    // ---- Phase A: load row (coalesced b128 clause), compute row max ------
    float4 v[Q_PER_LANE];
    float m = -INFINITY;
#pragma unroll
    for (int j = 0; j < Q_PER_LANE; ++j) {
        v[j] = xr[j * 32 + lane];
        m = fmaxf(m, fmaxf(fmaxf(v[j].x, v[j].y), fmaxf(v[j].z, v[j].w)));
    }
    m = wave_max(m);

    // am1z = 0.5 * (x - m), in place (matches reference: sub then exact *0.5)
#pragma unroll
    for (int j = 0; j < Q_PER_LANE; ++j) {
        v[j].x = (v[j].x - m) * 0.5f;
        v[j].y = (v[j].y - m) * 0.5f;
        v[j].z = (v[j].z - m) * 0.5f;
        v[j].w = (v[j].w - m) * 0.5f;
    }

    // ---- Phase B: ballot-compact candidates (am1z > -1) into LDS ---------
    // Pre-fill first 128 slots with PADVAL so register-chunk reads are valid.
#pragma unroll
    for (int k = 0; k < 4; ++k) my[lane + k * 32] = PADVAL;

    const unsigned lt = (1u << lane) - 1u;
    int cnt = 0;
    // Branchless emit: non-candidates store to the trash slot (one cndmask,
    // no EXEC save/restore churn).
    auto emit = [&](float val) {
        bool pred = val > -1.0f;
        unsigned mask = (unsigned)__ballot(pred);
        int idx = pred ? (cnt + __popc(mask & lt)) : trash;
        my[idx] = val;
        cnt += __popc(mask);
    };
#pragma unroll
    for (int j = 0; j < Q_PER_LANE; ++j) {
        emit(v[j].x); emit(v[j].y); emit(v[j].z); emit(v[j].w);
    }

    // Pad to a multiple of 32 (max element guarantees cnt >= 1), branchless.
    const int padded = (cnt + 31) & ~31;
    my[(lane < padded - cnt) ? (cnt + lane) : trash] = PADVAL;
    __syncthreads();   // single point; orders DS stores -> loads conservatively

    // First 128 candidates live in registers: zero LDS traffic per iteration
    // in the common case (~55 candidates/row for Gaussian data).
    const float r0 = my[lane];
    const float r1 = my[lane + 32];
    const float r2 = my[lane + 64];
    const float r3 = my[lane + 96];

    // ---- Phase C/D: bisection + normalization, specialized on chunk count
    float2 ti;
    if (padded <= 32) {
        ti = bisect_row<1, false>(my, r0, r1, r2, r3, padded, lane);
    } else if (padded <= 64) {
        ti = bisect_row<2, false>(my, r0, r1, r2, r3, padded, lane);
    } else if (padded <= 128) {
        ti = bisect_row<4, false>(my, r0, r1, r2, r3, padded, lane);
    } else {
        ti = bisect_row<4, true>(my, r0, r1, r2, r3, padded, lane);
    }
    const float tau = ti.x;
    const float inv = ti.y;

    // ---- Phase E: emit outputs from registers (b128 store clause) --------
#pragma unroll
    for (int j = 0; j < Q_PER_LANE; ++j) {
        float4 o;
        float t;
        t = fmaxf(v[j].x - tau, 0.0f); o.x = t * t * inv;
        t = fmaxf(v[j].y - tau, 0.0f); o.y = t * t * inv;
        t = fmaxf(v[j].z - tau, 0.0f); o.z = t * t * inv;
        t = fmaxf(v[j].w - tau, 0.0f); o.w = t * t * inv;
        orow[j * 32 + lane] = o;
    }
}

extern "C" void kernel_launch(void* const* d_in, const int* in_sizes, int n_in,
                              void* d_out, int out_size, void* d_ws, size_t ws_size,
                              hipStream_t stream) {
    (void)n_in; (void)d_ws; (void)ws_size; (void)out_size;
    const float* x = (const float*)d_in[0];
    float* out = (float*)d_out;
    const int nrows = in_sizes[0] / ROWLEN;   // 8 * 2048 = 16384
    const int blocks = (nrows + WAVES_PER_BLOCK - 1) / WAVES_PER_BLOCK;
    entmax15_rows_kernel<<<dim3(blocks), dim3(32 * WAVES_PER_BLOCK), 0, stream>>>(x, out, nrows);
}